// LightGCN_xij_5239860101649
// MI455X (gfx1250) — compile-verified
//
#include <hip/hip_runtime.h>

// ---------------------------------------------------------------------------
// LightGCN-xij for MI455X (gfx1250, wave32).
//  - Propagation: f32 atomic scatter into L2-resident [N x 64] buffers.
//  - Scoring: 16 rows per wave; softmax reductions via __shfl_xor(16);
//    final sum(u * softmax(i)) via V_WMMA_F32_16X16X4_F32 with ones-B
//    (matrix-core row reduction, fully f32).
// ---------------------------------------------------------------------------

typedef __attribute__((ext_vector_type(2))) float v2f;
typedef __attribute__((ext_vector_type(8))) float v8f;

#define D_DIM 64

// cur = concat(user_table, item_table); acc = same
__global__ void lgcn_init_kernel(const float* __restrict__ ut,
                                 const float* __restrict__ it,
                                 float* __restrict__ cur,
                                 float* __restrict__ acc,
                                 long nUserElems, long total) {
  long i = (long)blockIdx.x * blockDim.x + threadIdx.x;
  if (i >= total) return;
  float v = (i < nUserElems) ? ut[i] : it[i - nUserElems];
  cur[i] = v;
  acc[i] = v;
}

__global__ void lgcn_zero_kernel(float* __restrict__ p, long n) {
  long i = (long)blockIdx.x * blockDim.x + threadIdx.x;
  if (i < n) p[i] = 0.0f;
}

// acc += nxt; and zero zbuf (the buffer that becomes the next scatter target)
__global__ void lgcn_addzero_kernel(float* __restrict__ acc,
                                    const float* __restrict__ nxt,
                                    float* __restrict__ zbuf, long n) {
  long i = (long)blockIdx.x * blockDim.x + threadIdx.x;
  if (i >= n) return;
  acc[i] += nxt[i];
  zbuf[i] = 0.0f;
}

// One SpMM hop: nxt[row] += val * cur[col], 16 threads per edge (4 dims each).
__global__ void lgcn_scatter_kernel(const float* __restrict__ cur,
                                    float* __restrict__ nxt,
                                    const int* __restrict__ er,
                                    const int* __restrict__ ec,
                                    const float* __restrict__ ev, int E) {
  long t = (long)blockIdx.x * blockDim.x + threadIdx.x;
  int e = (int)(t >> 4);
  int j = (int)(t & 15);
  if (e >= E) return;
  const int r = er[e];
  const int c = ec[e];
  const float v = ev[e];
  const float4 x = reinterpret_cast<const float4*>(cur + (size_t)c * D_DIM)[j];
  float* dst = nxt + (size_t)r * D_DIM + 4 * j;
  atomicAdd(dst + 0, v * x.x);
  atomicAdd(dst + 1, v * x.y);
  atomicAdd(dst + 2, v * x.z);
  atomicAdd(dst + 3, v * x.w);
}

// Scoring: one wave handles 16 batch rows; 72-wide concat dim split across
// lane halves per the WMMA 16x16x4 f32 A-matrix layout:
//   lane L (<16):  K = {4c+0, 4c+1} of chunk c, row M = L
//   lane L+16:     K = {4c+2, 4c+3} of chunk c, row M = L
__global__ void lgcn_score_kernel(const float* __restrict__ acc,
                                  const int* __restrict__ users,
                                  const int* __restrict__ items,
                                  const float* __restrict__ xij,
                                  const float* __restrict__ xij_table,
                                  float* __restrict__ out,
                                  int nUser, int B) {
  const int lane = threadIdx.x & 31;
  const int wave = threadIdx.x >> 5;
  const int half = lane >> 4;   // which K pair of each 4-chunk
  const int m    = lane & 15;   // row within 16-row tile
  const int rowBase = (blockIdx.x * (blockDim.x >> 5) + wave) * 16;
  const int row = rowBase + m;  // grid sized so row < B always

  const float* ubase = acc + (size_t)users[row] * D_DIM;
  const float* ibase = acc + ((size_t)nUser + (size_t)items[row]) * D_DIM;
  const float xv = xij[row] - 0.3f;

  // Each lane owns 36 of the 72 concat positions (chunks 0..17, 2 each).
  float zu[36], zi[36];
  float mx = -3.402823466e38f;
#pragma unroll
  for (int c = 0; c < 18; ++c) {
#pragma unroll
    for (int j = 0; j < 2; ++j) {
      const int k = 4 * c + 2 * half + j;
      float a, b;
      if (4 * c < 64) {                 // compile-time split per chunk
        a = 0.25f * ubase[k];           // light = acc / (L+1), L=3
        b = 0.25f * ibase[k];
      } else {
        const float xe = xij_table[k - 64] * xv;
        a = xe;
        b = xe;
      }
      zu[2 * c + j] = a;
      zi[2 * c + j] = b;
      mx = fmaxf(mx, b);
    }
  }
  // combine the two half-lanes of each row (wave32 shuffle)
  mx = fmaxf(mx, __shfl_xor(mx, 16, 32));

  float s = 0.0f;
#pragma unroll
  for (int t = 0; t < 36; ++t) s += expf(zi[t] - mx);
  s += __shfl_xor(s, 16, 32);
  const float inv_s = 1.0f / s;

#if __has_builtin(__builtin_amdgcn_wmma_f32_16x16x4_f32)
  // Row-sum via matrix core: D = A(16x4 products) x B(4x16 ones) + C.
  v8f cacc = {0.0f, 0.0f, 0.0f, 0.0f, 0.0f, 0.0f, 0.0f, 0.0f};
  const v2f bones = {1.0f, 1.0f};
#pragma unroll
  for (int c = 0; c < 18; ++c) {
    v2f a;
#pragma unroll
    for (int j = 0; j < 2; ++j) {
      const float u = 1.0f / (1.0f + expf(-zu[2 * c + j]));
      const float p = expf(zi[2 * c + j] - mx) * inv_s;
      a[j] = u * p;
    }
    cacc = __builtin_amdgcn_wmma_f32_16x16x4_f32(
        false, a, false, bones, (short)0, cacc, false, false);
  }
  // D layout: VGPR r -> {lanes 0-15: M=r, N=lane} {lanes 16-31: M=8+r}.
  // Every N column holds the same row sum; lane 0 emits rows 0-7,
  // lane 16 emits rows 8-15.
  if (m == 0) {
#pragma unroll
    for (int r = 0; r < 8; ++r) out[rowBase + half * 8 + r] = cacc[r];
  }
#else
  // Fallback: per-lane partial sum + half-lane combine.
  float ps = 0.0f;
#pragma unroll
  for (int t = 0; t < 36; ++t) {
    const float u = 1.0f / (1.0f + expf(-zu[t]));
    const float p = expf(zi[t] - mx) * inv_s;
    ps += u * p;
  }
  ps += __shfl_xor(ps, 16, 32);
  if (half == 0) out[row] = ps;
#endif
}

extern "C" void kernel_launch(void* const* d_in, const int* in_sizes, int n_in,
                              void* d_out, int out_size, void* d_ws, size_t ws_size,
                              hipStream_t stream) {
  const int B = in_sizes[0];
  const int E = in_sizes[6];
  const long NU = (long)in_sizes[3] / D_DIM;
  const long NI = (long)in_sizes[4] / D_DIM;
  const long N = NU + NI;
  const long ND = N * D_DIM;

  const int*   users = (const int*)d_in[0];
  const int*   items = (const int*)d_in[1];
  const float* xij   = (const float*)d_in[2];
  const float* ut    = (const float*)d_in[3];
  const float* it    = (const float*)d_in[4];
  const float* xt    = (const float*)d_in[5];
  const int*   er    = (const int*)d_in[6];
  const int*   ec    = (const int*)d_in[7];
  const float* ev    = (const float*)d_in[8];
  float* out = (float*)d_out;

  // Workspace layout: [cur | next | acc], each N*64 f32 (3 x 38.4 MB).
  float* bufA = (float*)d_ws;
  float* bufB = bufA + ND;
  float* accb = bufA + 2 * ND;

  const dim3 blk(256);
  const int gElem = (int)((ND + 255) / 256);

  lgcn_init_kernel<<<gElem, blk, 0, stream>>>(ut, it, bufA, accb, NU * D_DIM, ND);
  lgcn_zero_kernel<<<gElem, blk, 0, stream>>>(bufB, ND);

  float* cur = bufA;
  float* nxt = bufB;
  const int gScat = (int)(((long)E * 16 + 255) / 256);
  for (int l = 0; l < 3; ++l) {
    lgcn_scatter_kernel<<<gScat, blk, 0, stream>>>(cur, nxt, er, ec, ev, E);
    lgcn_addzero_kernel<<<gElem, blk, 0, stream>>>(accb, nxt, cur, ND);
    float* t = cur; cur = nxt; nxt = t;
  }

  // 16 rows per wave, 8 waves per 256-thread block -> 128 rows per block.
  lgcn_score_kernel<<<B / 128, blk, 0, stream>>>(accb, users, items, xij, xt,
                                                 out, (int)NU, B);
}